// CAFBlock_45303315038549
// MI455X (gfx1250) — compile-verified
//
#include <hip/hip_runtime.h>
#include <cstdint>

// ---------------------------------------------------------------------------
// CAF block, MI455X (gfx1250). Bandwidth-bound (~402 MB traffic, ~0.5 GFLOP).
// CDNA5 paths used:
//   * v_wmma_f32_16x16x4_f32  : wave-level sum reductions (ones-matrix trick)
//   * global_load_async_to_lds_b128 + s_wait_asynccnt : double-buffered
//     staging pipeline in the fused streaming pass (ASYNCcnt)
//   * global_prefetch_b8, NT b128 stores
// ---------------------------------------------------------------------------

typedef float v2f __attribute__((ext_vector_type(2)));
typedef float v4f __attribute__((ext_vector_type(4)));
typedef float v8f __attribute__((ext_vector_type(8)));

#define Bsz   2
#define Csz   512
#define Tsz   256
#define Fsz   128
#define TVsz  64
#define SLAB  (Tsz * Fsz)          // 32768 floats per (b,c)
#define NBTF  (Bsz * Tsz * Fsz)    // 65536 elems per channel (BN stats)
#define NCT   (Csz * TVsz)         // 32768 elems per sample (GN stats)
#define EPSV  1e-5f

// Wave-level sum of one float per lane via V_WMMA_F32_16X16X4_F32.
// A layout (32-bit, 16x4): lane<16 holds A[l][0..1], lane>=16 holds A[l-16][2..3].
// Put x in both A VGPRs, B = all-ones (layout-invariant):
//   D[m,n] = 2*(x_m + x_{m+16})  for every n.
// Lane l<16 holds column n=l, rows 0..7 in its 8 C VGPRs; lanes>=16 rows 8..15.
// So sum-of-8-VGPRs + shfl_xor(16) gives 2 * (wave sum) in every lane.
__device__ __forceinline__ float wave_sum_wmma(float x) {
    v2f a;  a.x = x;  a.y = x;
    v2f ones; ones.x = 1.f; ones.y = 1.f;
    v8f c = {};
    c = __builtin_amdgcn_wmma_f32_16x16x4_f32(false, a, false, ones,
                                              (short)0, c, false, false);
    float t = c[0] + c[1] + c[2] + c[3] + c[4] + c[5] + c[6] + c[7];
    t += __shfl_xor(t, 16);
    return t * 0.5f;
}

// ---- Kernel 1: per-channel raw moments of audio (serves BOTH BN branches) --
__global__ void k_bn_stats(const float* __restrict__ audio,
                           float* __restrict__ sums,
                           float* __restrict__ sumsq) {
    const int c   = blockIdx.x;     // 0..511
    const int tid = threadIdx.x;    // 0..255
    float s = 0.f, q = 0.f;
    for (int b = 0; b < Bsz; ++b) {
        const v4f* p = (const v4f*)(audio + (size_t)(b * Csz + c) * SLAB);
        // 32768 floats = 8192 float4, 256 threads -> 32 iters, coalesced b128
        for (int i = tid; i < SLAB / 4; i += 256) {
            __builtin_prefetch(p + i + 2048, 0, 1);  // global_prefetch_b8, ~32KB ahead
            v4f v = p[i];
            s += v.x + v.y + v.z + v.w;
            q += v.x * v.x + v.y * v.y + v.z * v.z + v.w * v.w;
        }
    }
    // wave-level reduce via WMMA, then tiny cross-wave combine (8 waves)
    const float ws = wave_sum_wmma(s);
    const float wq = wave_sum_wmma(q);
    __shared__ float rs[8], rq[8];
    const int wid = tid >> 5, lane = tid & 31;
    if (lane == 0) { rs[wid] = ws; rq[wid] = wq; }
    __syncthreads();
    if (tid == 0) {
        float S = 0.f, Q = 0.f;
        for (int w = 0; w < 8; ++w) { S += rs[w]; Q += rq[w]; }
        sums[c] = S; sumsq[c] = Q;
    }
}

// ---- Kernel 2: per-sample moments of the two affine-transformed videos ----
__global__ void k_vid_stats(const float* __restrict__ video,
                            const float* __restrict__ aw, const float* __restrict__ ab,
                            const float* __restrict__ rw, const float* __restrict__ rb,
                            float* __restrict__ vstats /* [B][4] */) {
    const int b = blockIdx.x, tid = threadIdx.x;
    const float* base = video + (size_t)b * NCT;
    float s1a = 0.f, s2a = 0.f, s1r = 0.f, s2r = 0.f;
    for (int i = tid; i < NCT; i += 256) {
        const int c = i >> 6;          // channel of this element
        const float x  = base[i];
        const float ya = x * aw[c] + ab[c];
        const float yr = x * rw[c] + rb[c];
        s1a += ya; s2a += ya * ya; s1r += yr; s2r += yr * yr;
    }
    const float w0 = wave_sum_wmma(s1a);
    const float w1 = wave_sum_wmma(s2a);
    const float w2 = wave_sum_wmma(s1r);
    const float w3 = wave_sum_wmma(s2r);
    __shared__ float r[4][8];
    const int wid = tid >> 5, lane = tid & 31;
    if (lane == 0) { r[0][wid] = w0; r[1][wid] = w1; r[2][wid] = w2; r[3][wid] = w3; }
    __syncthreads();
    if (tid < 4) {
        float S = 0.f;
        for (int w = 0; w < 8; ++w) S += r[tid][w];
        vstats[b * 4 + tid] = S;
    }
}

// ---- Kernel 3: fold depthwise-1x1 + BN into per-channel affine (A,B) ------
// BN(x*w) = x*A + B, A = w*gamma*rsqrt(w^2*var + eps), B = beta - mean*A
__global__ void k_bn_coeff(const float* __restrict__ sums, const float* __restrict__ sumsq,
                           const float* __restrict__ vw, const float* __restrict__ vg,
                           const float* __restrict__ vb,
                           const float* __restrict__ gw, const float* __restrict__ gg,
                           const float* __restrict__ gb,
                           float* __restrict__ coeff /* [4][512] */) {
    const int c = blockIdx.x * blockDim.x + threadIdx.x;
    if (c >= Csz) return;
    const float invN = 1.f / (float)NBTF;
    const float m   = sums[c] * invN;
    const float var = sumsq[c] * invN - m * m;   // biased, matches jnp.var
    const float Av = vw[c] * vg[c] * rsqrtf(vw[c] * vw[c] * var + EPSV);
    const float Ag = gw[c] * gg[c] * rsqrtf(gw[c] * gw[c] * var + EPSV);
    coeff[c]            = Av;
    coeff[Csz + c]      = vb[c] - m * Av;
    coeff[2 * Csz + c]  = Ag;
    coeff[3 * Csz + c]  = gb[c] - m * Ag;
}

// ---- Kernel 4: GN + softmax(Tv) per (b,c); emit att[B,C,Tv], vr[B,C,Tv] ---
__global__ void k_vid_norm(const float* __restrict__ video,
                           const float* __restrict__ vstats,
                           const float* __restrict__ aw, const float* __restrict__ ab,
                           const float* __restrict__ agam, const float* __restrict__ abet,
                           const float* __restrict__ rw, const float* __restrict__ rb,
                           const float* __restrict__ rgam, const float* __restrict__ rbet,
                           float* __restrict__ att_out, float* __restrict__ vr_out) {
    const int bc = blockIdx.x;            // 0..1023
    const int b  = bc >> 9, c = bc & (Csz - 1);
    const int tv = threadIdx.x;           // 0..63
    const float invN = 1.f / (float)NCT;
    const float m_a = vstats[b * 4 + 0] * invN;
    const float v_a = vstats[b * 4 + 1] * invN - m_a * m_a;
    const float m_r = vstats[b * 4 + 2] * invN;
    const float v_r = vstats[b * 4 + 3] * invN - m_r * m_r;
    const float ia = rsqrtf(v_a + EPSV), ir = rsqrtf(v_r + EPSV);

    const float x  = video[((size_t)b * Csz + c) * TVsz + tv];
    const float ya = ((x * aw[c] + ab[c]) - m_a) * ia * agam[c] + abet[c];
    const float yr = ((x * rw[c] + rb[c]) - m_r) * ir * rgam[c] + rbet[c];

    __shared__ float sm[TVsz];
    sm[tv] = ya; __syncthreads();
    for (int off = 32; off > 0; off >>= 1) {
        if (tv < off) sm[tv] = fmaxf(sm[tv], sm[tv + off]);
        __syncthreads();
    }
    const float mx = sm[0];
    __syncthreads();
    const float e = __expf(ya - mx);
    sm[tv] = e; __syncthreads();
    for (int off = 32; off > 0; off >>= 1) {
        if (tv < off) sm[tv] += sm[tv + off];
        __syncthreads();
    }
    const float inv_sum = 1.f / sm[0];
    att_out[(size_t)bc * TVsz + tv] = e * inv_sum;
    vr_out [(size_t)bc * TVsz + tv] = yr;
}

// ---- Kernel 5: fused streaming pass, double-buffered async global->LDS ----
// out = att[t>>2]*(x*Av+Bv) + vr[t>>2]*relu(x*Ag+Bg)
__global__ void k_fuse(const float* __restrict__ audio,
                       const float* __restrict__ coeff,
                       const float* __restrict__ att,
                       const float* __restrict__ vr,
                       float* __restrict__ out) {
    __shared__ float lds_att[TVsz];
    __shared__ float lds_vr[TVsz];
    __shared__ __align__(16) float tile[2][1024];   // 2 x 4 KB staging buffers

    const int bc  = blockIdx.x;           // (b*512 + c), 0..1023
    const int c   = bc & (Csz - 1);
    const int tid = threadIdx.x;          // 0..255

    if (tid < TVsz) {
        lds_att[tid] = att[(size_t)bc * TVsz + tid];
        lds_vr[tid]  = vr [(size_t)bc * TVsz + tid];
    }
    const float Av = coeff[c];
    const float Bv = coeff[Csz + c];
    const float Ag = coeff[2 * Csz + c];
    const float Bg = coeff[3 * Csz + c];
    __syncthreads();

    const float* gbase = audio + (size_t)bc * SLAB;
    float*       obase = out   + (size_t)bc * SLAB;

    // Low 32 bits of a flat pointer to a __shared__ object == LDS byte offset
    // (CDNA5 aperture model: the LDS address is the low 32 bits).
    const unsigned lds_base = (unsigned)(uintptr_t)(&tile[0][0]);
    const int iters = SLAB / 1024;        // 32 tiles of 4 KB per block

    // Prologue: stage tile 0 (each lane stages & later consumes its own 16 B,
    // so no cross-wave sharing -> no barriers needed in the pipeline).
    {
        unsigned ldsoff = lds_base + tid * 16u;
        const float* g  = gbase + tid * 4;
        asm volatile("global_load_async_to_lds_b128 %0, %1, off"
                     :: "v"(ldsoff), "v"(g) : "memory");
    }

    for (int i = 0; i < iters; ++i) {
        const int buf = i & 1;
        if (i + 1 < iters) {
            unsigned ldsoff = lds_base + (unsigned)((i + 1) & 1) * 4096u + tid * 16u;
            const float* g  = gbase + (i + 1) * 1024 + tid * 4;
            asm volatile("global_load_async_to_lds_b128 %0, %1, off"
                         :: "v"(ldsoff), "v"(g) : "memory");
            asm volatile("s_wait_asynccnt 0x1" ::: "memory");  // tile i landed
        } else {
            asm volatile("s_wait_asynccnt 0x0" ::: "memory");
        }

        const int p  = i * 1024 + tid * 4;   // float offset in slab
        const int tv = p >> 9;               // t = p>>7; nearest-resize: t>>2
        const float a  = lds_att[tv];
        const float gv = lds_vr[tv];

        v4f x   = *(const v4f*)&tile[buf][tid * 4];
        v4f val = x * Av + Bv;
        v4f gt  = x * Ag + Bg;
        gt      = __builtin_elementwise_max(gt, (v4f)0.f);   // ReLU
        v4f o   = a * val + gv * gt;
        __builtin_nontemporal_store(o, (v4f*)(obase + p));   // streamed output
    }
}

// ---------------------------------------------------------------------------
extern "C" void kernel_launch(void* const* d_in, const int* in_sizes, int n_in,
                              void* d_out, int out_size, void* d_ws, size_t ws_size,
                              hipStream_t stream) {
    const float* audio   = (const float*)d_in[0];
    const float* video   = (const float*)d_in[1];
    const float* value_w = (const float*)d_in[2];
    const float* value_g = (const float*)d_in[3];
    const float* value_b = (const float*)d_in[4];
    const float* gate_w  = (const float*)d_in[5];
    const float* gate_g  = (const float*)d_in[6];
    const float* gate_b  = (const float*)d_in[7];
    const float* att_w   = (const float*)d_in[8];
    const float* att_b   = (const float*)d_in[9];
    const float* att_g   = (const float*)d_in[10];
    const float* att_bt  = (const float*)d_in[11];
    const float* res_w   = (const float*)d_in[12];
    const float* res_b   = (const float*)d_in[13];
    const float* res_g   = (const float*)d_in[14];
    const float* res_bt  = (const float*)d_in[15];
    float* outp = (float*)d_out;

    // Workspace layout (floats): 512 + 512 + 8 + 2048 + 65536 + 65536 ~= 537 KB
    float* ws     = (float*)d_ws;
    float* sums   = ws;
    float* sumsq  = ws + Csz;
    float* vstats = ws + 2 * Csz;
    float* coeff  = ws + 2 * Csz + 8;
    float* attbuf = coeff + 4 * Csz;
    float* vrbuf  = attbuf + (size_t)Bsz * Csz * TVsz;

    k_bn_stats<<<Csz, 256, 0, stream>>>(audio, sums, sumsq);
    k_vid_stats<<<Bsz, 256, 0, stream>>>(video, att_w, att_b, res_w, res_b, vstats);
    k_bn_coeff<<<2, 256, 0, stream>>>(sums, sumsq, value_w, value_g, value_b,
                                      gate_w, gate_g, gate_b, coeff);
    k_vid_norm<<<Bsz * Csz, TVsz, 0, stream>>>(video, vstats,
                                               att_w, att_b, att_g, att_bt,
                                               res_w, res_b, res_g, res_bt,
                                               attbuf, vrbuf);
    k_fuse<<<Bsz * Csz, 256, 0, stream>>>(audio, coeff, attbuf, vrbuf, outp);
}